// HogLayer3D_7052336300381
// MI455X (gfx1250) — compile-verified
//
#include <hip/hip_runtime.h>
#include <math.h>

typedef float v2f __attribute__((ext_vector_type(2)));
typedef float v8f __attribute__((ext_vector_type(8)));

#define CELL   7
#define BINS   20
#define HDIM   112
#define NCELL  16          // cells per spatial dim
#define NBLK   15          // blocks per spatial dim
#define DSEG   56          // d-extent handled per workgroup (8 cells)
#define CSEG   8           // cells along d per workgroup
#define TI     9
#define TJ     9
#define TK     (DSEG + 2)  // 58
#define TILE_N (TI * TJ * TK)          // 4698
#define NVOX   (CELL * CELL * DSEG)    // 2744
#define NSTEP  ((NVOX + 255) / 256)    // 11

// 20 dodecahedron vertices (rows), 3 components each
#define PHI  1.6180339887498949f
#define RPHI 0.6180339887498949f
__constant__ float c_vert[20][3] = {
    { 1.f,  1.f,  1.f}, {-1.f, -1.f, -1.f}, { 1.f,  1.f, -1.f}, {-1.f, -1.f,  1.f},
    { 1.f, -1.f,  1.f}, {-1.f,  1.f, -1.f}, { 1.f, -1.f, -1.f}, {-1.f,  1.f,  1.f},
    { 0.f,  RPHI, PHI}, { 0.f, -RPHI, -PHI}, { 0.f,  RPHI, -PHI}, { 0.f, -RPHI, PHI},
    { PHI, 0.f,  RPHI}, {-PHI, 0.f, -RPHI}, { PHI, 0.f, -RPHI}, {-PHI, 0.f,  RPHI},
    { RPHI, PHI, 0.f}, {-RPHI, -PHI, 0.f}, { RPHI, -PHI, 0.f}, {-RPHI, PHI, 0.f}};

__device__ __forceinline__ int mirror(int c) {
    c = (c < 0) ? -c : c;
    return (c > HDIM - 1) ? (2 * (HDIM - 1) - c) : c;
}

// ds_swizzle SWAPX16 (xor=0x10, and=0x1f): exchange wave halves
__device__ __forceinline__ float swap16f(float x) {
    return __int_as_float(__builtin_amdgcn_ds_swizzle(__float_as_int(x), 0x401F));
}
__device__ __forceinline__ int swap16i(int x) {
    return __builtin_amdgcn_ds_swizzle(x, 0x401F);
}

// Lexicographic (max value, min bin) combine — branch-free (no short-circuit ops),
// total order => any reduction tree is valid and matches argmax first-max tie-break.
__device__ __forceinline__ void amax2(float& v1, int& b1, float v2, int b2) {
    bool take = (v2 > v1) | ((v2 == v1) & (b2 < b1));
    v1 = take ? v2 : v1;
    b1 = take ? b2 : b1;
}

// Reduce the 20 projections for one voxel. Lane v holds bins 0-7 (d1) and 16-19
// (d2, low half only); lane v+16 holds bins 8-15 (d1). One SWAPX16 merges halves.
__device__ __forceinline__ int argmax_bins(v8f d1, v8f d2, bool lowHalf) {
    const float NEG = -__builtin_inff();
    const int   bb0 = lowHalf ? 0 : 8;
    float cv[12];
    int   cb[12];
#pragma unroll
    for (int r = 0; r < 8; ++r) { cv[r] = d1[r]; cb[r] = bb0 + r; }
#pragma unroll
    for (int r = 0; r < 4; ++r) { cv[8 + r] = lowHalf ? d2[r] : NEG; cb[8 + r] = 16 + r; }
    // tree: 12 -> 6 -> 3 -> 1 (depth 4 instead of a 12-deep serial chain)
#pragma unroll
    for (int s = 0; s < 6; ++s) amax2(cv[s], cb[s], cv[6 + s], cb[6 + s]);
    amax2(cv[0], cb[0], cv[3], cb[3]);
    amax2(cv[1], cb[1], cv[4], cb[4]);
    amax2(cv[2], cb[2], cv[5], cb[5]);
    amax2(cv[0], cb[0], cv[1], cb[1]);
    amax2(cv[0], cb[0], cv[2], cb[2]);
    // cross-half combine
    float ov = swap16f(cv[0]);
    int   oi = swap16i(cb[0]);
    amax2(cv[0], cb[0], ov, oi);
    return cb[0];
}

// ---------------------------------------------------------------------------
// Kernel 1: fused Sobel + WMMA phase projection + argmax + cell histograms.
// Grid: (b=2) x (hc=16) x (wc=16) x (seg=2) = 1024 workgroups of 256 threads.
// ---------------------------------------------------------------------------
__global__ __launch_bounds__(256) void hog3d_hist_kernel(
    const float* __restrict__ x, float* __restrict__ hist) {
    __shared__ float tile[TILE_N];
    __shared__ float s_hist[CSEG * BINS];  // 8 cells * 20 bins

    const int blk = blockIdx.x;
    const int b   = blk >> 9;
    const int rem = blk & 511;
    const int hc  = rem >> 5;
    const int wc  = (rem >> 1) & 15;
    const int seg = rem & 1;
    const int tid = threadIdx.x;

    for (int t = tid; t < CSEG * BINS; t += 256) s_hist[t] = 0.f;

    // --- stage input tile (with reflect halo) into LDS ---
    const int h0 = hc * CELL, w0 = wc * CELL, d0 = seg * DSEG;
    const int xbase = b * HDIM * HDIM * HDIM;
    for (int t = tid; t < TILE_N; t += 256) {
        int i = t / (TJ * TK);
        int r = t - i * (TJ * TK);
        int j = r / TK;
        int k = r - j * TK;
        int gi = mirror(h0 - 1 + i);
        int gj = mirror(w0 - 1 + j);
        int gk = mirror(d0 - 1 + k);
        tile[t] = x[xbase + (gi * HDIM + gj) * HDIM + gk];
    }
    __syncthreads();

    const int  lane    = tid & 31;
    const bool lowHalf = lane < 16;
    const int  m       = lane & 15;

    // Loop-invariant A fragments (16x4 f32 layout: lanes 0-15 K0/K1, 16-31 K2/K3)
    v2f a1, a2;
    a1.x = lowHalf ? c_vert[m][0] : c_vert[m][2];
    a1.y = lowHalf ? c_vert[m][1] : 0.f;
    a2.x = (m < 4) ? (lowHalf ? c_vert[16 + m][0] : c_vert[16 + m][2]) : 0.f;
    a2.y = (m < 4 && lowHalf) ? c_vert[16 + m][1] : 0.f;

    for (int step = 0; step < NSTEP; ++step) {
        int  vraw   = step * 256 + tid;
        bool active = vraw < NVOX;
        int  v      = active ? vraw : (NVOX - 1);
        int  dl     = v % DSEG;
        int  t2     = v / DSEG;
        int  wl     = t2 % CELL;
        int  hl     = t2 / CELL;

        // --- separable 3x3x3 Sobel from LDS ---
        float u[3][3], ud[3][3];
#pragma unroll
        for (int a = 0; a < 3; ++a)
#pragma unroll
            for (int bb = 0; bb < 3; ++bb) {
                int   base = ((hl + a) * TJ + (wl + bb)) * TK + dl;
                float t0 = tile[base], t1 = tile[base + 1], t2v = tile[base + 2];
                u[a][bb]  = t0 + 2.f * t1 + t2v;   // smooth along d
                ud[a][bb] = t2v - t0;              // diff along d
            }
        float gh = (u[2][0] + 2.f * u[2][1] + u[2][2]) - (u[0][0] + 2.f * u[0][1] + u[0][2]);
        float gw = (u[0][2] + 2.f * u[1][2] + u[2][2]) - (u[0][0] + 2.f * u[1][0] + u[2][0]);
        float gd = (ud[0][0] + 2.f * ud[0][1] + ud[0][2])
                 + 2.f * (ud[1][0] + 2.f * ud[1][1] + ud[1][2])
                 + (ud[2][0] + 2.f * ud[2][1] + ud[2][2]);
        float nrm = sqrtf(gh * gh + gw * gw + gd * gd);

        // --- B fragments: voxel-per-lane, replicate across halves via SWAPX16 ---
        float sgh = swap16f(gh), sgw = swap16f(gw), sgd = swap16f(gd);
        v2f b1, b2;
        b1.x = lowHalf ? gh : sgd;   // K0 / K2
        b1.y = lowHalf ? gw : 0.f;   // K1 / K3(=0)
        b2.x = lowHalf ? sgh : gd;
        b2.y = lowHalf ? sgw : 0.f;

        v8f z = {0.f, 0.f, 0.f, 0.f, 0.f, 0.f, 0.f, 0.f};
        // D = vertices x gradients: bins (M) across VGPRs, voxels (N) across lanes
        v8f d1a = __builtin_amdgcn_wmma_f32_16x16x4_f32(false, a1, false, b1, (short)0, z, false, false);
        v8f d2a = __builtin_amdgcn_wmma_f32_16x16x4_f32(false, a2, false, b1, (short)0, z, false, false);
        v8f d1b = __builtin_amdgcn_wmma_f32_16x16x4_f32(false, a1, false, b2, (short)0, z, false, false);
        v8f d2b = __builtin_amdgcn_wmma_f32_16x16x4_f32(false, a2, false, b2, (short)0, z, false, false);

        int binA = argmax_bins(d1a, d2a, lowHalf);  // voxels 0..15 of this wave
        int binB = argmax_bins(d1b, d2b, lowHalf);  // voxels 16..31 of this wave

        int bin   = lowHalf ? binA : binB;  // each lane keeps its own voxel's bin
        int dcell = dl / CELL;              // 0..7
        atomicAdd(&s_hist[dcell * BINS + bin], active ? nrm : 0.f);
    }
    __syncthreads();

    // write 8 cells x 20 bins; layout hist[b][hc][wc][dc][bin]
    for (int t = tid; t < CSEG * BINS; t += 256) {
        int dc_local = t / BINS;
        int bin      = t - dc_local * BINS;
        int dc       = seg * CSEG + dc_local;
        hist[(((b * NCELL + hc) * NCELL + wc) * NCELL + dc) * BINS + bin] = s_hist[t];
    }
}

// ---------------------------------------------------------------------------
// Kernel 2: 2x2x2 block gather + L2 normalize. One wave per 160-vector.
// ---------------------------------------------------------------------------
__global__ __launch_bounds__(256) void hog3d_block_kernel(
    const float* __restrict__ hist, float* __restrict__ out) {
    const int wave = threadIdx.x >> 5;
    const int lane = threadIdx.x & 31;
    const int site = blockIdx.x * 8 + wave;
    const int NSITE = 2 * NBLK * NBLK * NBLK;  // 6750
    if (site >= NSITE) return;

    int db = site % NBLK;
    int t  = site / NBLK;
    int wb = t % NBLK;
    t /= NBLK;
    int hb = t % NBLK;
    int b  = t / NBLK;

    float vals[5];
    float ss = 0.f;
#pragma unroll
    for (int q = 0; q < 5; ++q) {
        int idx = lane + 32 * q;        // 0..159 ; layout = bin*8 + (i*4+j*2+k)
        int bin = idx >> 3;
        int off = idx & 7;
        int i = off >> 2, j = (off >> 1) & 1, k = off & 1;
        int hidx = (((b * NCELL + hb + i) * NCELL + (wb + j)) * NCELL + (db + k)) * BINS + bin;
        float v = hist[hidx];
        vals[q] = v;
        ss += v * v;
    }
    // wave32 butterfly reduction via ds_swizzle (xor 1,2,4,8,16)
    ss += __int_as_float(__builtin_amdgcn_ds_swizzle(__float_as_int(ss), 0x041F));
    ss += __int_as_float(__builtin_amdgcn_ds_swizzle(__float_as_int(ss), 0x081F));
    ss += __int_as_float(__builtin_amdgcn_ds_swizzle(__float_as_int(ss), 0x101F));
    ss += __int_as_float(__builtin_amdgcn_ds_swizzle(__float_as_int(ss), 0x201F));
    ss += __int_as_float(__builtin_amdgcn_ds_swizzle(__float_as_int(ss), 0x401F));

    float inv = 1.f / fmaxf(sqrtf(ss), 1e-5f);
    int   obase = site * (BINS * 8);
#pragma unroll
    for (int q = 0; q < 5; ++q) out[obase + lane + 32 * q] = vals[q] * inv;
}

extern "C" void kernel_launch(void* const* d_in, const int* in_sizes, int n_in,
                              void* d_out, int out_size, void* d_ws, size_t ws_size,
                              hipStream_t stream) {
    (void)in_sizes; (void)n_in; (void)out_size; (void)ws_size;
    const float* x    = (const float*)d_in[0];
    float*       out  = (float*)d_out;
    float*       hist = (float*)d_ws;  // 2*16*16*16*20 f32 = 640 KB scratch

    // Kernel 1: (b=2) x (hc=16) x (wc=16) x (dseg=2) = 1024 blocks
    hog3d_hist_kernel<<<1024, 256, 0, stream>>>(x, hist);

    // Kernel 2: 6750 sites, 8 waves per block
    const int nsite = 2 * NBLK * NBLK * NBLK;
    hog3d_block_kernel<<<(nsite + 7) / 8, 256, 0, stream>>>(hist, out);
}